// IVFBook_36215164240450
// MI455X (gfx1250) — compile-verified
//
#include <hip/hip_runtime.h>

// VQ codebook search: dists = ||c||^2 - 2 x.c ; argmin over K; gather winners.
// B(M)=4096, D=256, K(N)=65536, fp32 via V_WMMA_F32_16X16X4_F32.
// B tiles staged through LDS by the Tensor Data Mover (double buffered),
// B fragments software-pipelined out of LDS (8-deep prefetch ring).

typedef float v2f __attribute__((ext_vector_type(2)));
typedef float v8f __attribute__((ext_vector_type(8)));
typedef unsigned int u32x4 __attribute__((ext_vector_type(4)));
typedef int i32x4 __attribute__((ext_vector_type(4)));
typedef int i32x8 __attribute__((ext_vector_type(8)));

#define M_TOT  4096
#define D_DIM  256
#define K_TOT  65536
#define NSPLIT 16                     // K split for parallelism
#define NCHUNK (K_TOT / NSPLIT)       // 4096 centroids per block
#define NITER  (NCHUNK / 16)          // 256 16-centroid tiles per block
#define WPB    4                      // waves per block (kernel 1)

#define TROW   260                    // LDS row stride in floats (TDM-padded, bank-conflict-free)
#define TILE_F (16 * TROW)            // floats per LDS tile
#define TILE_B (TILE_F * 4)           // bytes per LDS tile

#define PF     8                      // B-fragment prefetch depth (v2f regs)

// Tell the compiler the TDM wrote LDS: escape the pointer + memory clobber.
// Without this, clang sees no store to smem anywhere and folds all LDS loads
// to undef, deleting the whole WMMA loop (observed in round 2).
#define LDS_PUBLISH(p) asm volatile("" : : "r"(p) : "memory")

// ---------------------------------------------------------------------------
// TDM: DMA one 16x256 fp32 centroid tile (16 KB) into LDS with 16B row pad.
// D# per CDNA5 ISA ch.8. lds_off is the byte offset of the tile in LDS
// (single __shared__ allocation -> segment base 0).
// ---------------------------------------------------------------------------
__device__ __forceinline__ void tdm_load_tile(const float* gsrc, unsigned lds_off) {
  unsigned long long ga = (unsigned long long)(uintptr_t)gsrc;
  u32x4 g0 = { 1u,                                  // count=1, user descriptor
               lds_off,                             // lds_addr [63:32]
               (unsigned)ga,                        // global_addr [95:64]
               (unsigned)(ga >> 32) | 0x80000000u };// global_addr hi | type=2
  i32x8 g1 = { (int)0x07D20000,  // data_size=4B, pad_enable, pad_interval=256dw, pad_amount=4dw
               (int)0x01000000,  // tensor_dim0[15:0]=256 in bits 63:48
               (int)0x00100000,  // tensor_dim1[15:0]=16  in bits 95:80
               (int)0x01000000,  // tile_dim0=256 in bits 127:112
               16,               // tile_dim1=16
               256,              // tensor_dim0_stride=256
               0, 0 };
  i32x4 z4 = { 0, 0, 0, 0 };
#if __clang_major__ >= 23
  i32x8 z8 = { 0, 0, 0, 0, 0, 0, 0, 0 };
  __builtin_amdgcn_tensor_load_to_lds(g0, g1, z4, z4, z8, 0);
#else
  __builtin_amdgcn_tensor_load_to_lds(g0, g1, z4, z4, 0);
#endif
}

// ---------------------------------------------------------------------------
// Kernel 1: block = 4 waves x 16 rows (64 rows) vs one K-split of centroids.
// A (x tile) lives in 128 VGPRs per wave; B tiles arrive via TDM double buffer.
// Running (min,argmin) kept in the WMMA C/D lane layout; reduced at the end.
// ---------------------------------------------------------------------------
__global__ __launch_bounds__(128) void vq_argmin_partial(
    const float* __restrict__ x,       // [4096,256]
    const float* __restrict__ cent,    // [65536,256]
    float2* __restrict__ part)         // [4096, NSPLIT] (minval, idx-bits)
{
  __shared__ float smem[2 * TILE_F];   // 33,280 B: two TDM tiles

  const int lane  = threadIdx.x & 31;
  const int wave  = threadIdx.x >> 5;
  const int nspl  = blockIdx.x;                    // 0..NSPLIT-1
  const int mbase = (blockIdx.y * WPB + wave) * 16;
  const int nbase0 = nspl * NCHUNK;

  const int lrow  = lane & 15;        // A: row M ; B: col N
  const int khalf = (lane >> 4) * 2;  // half-wave K offset (0 or 2)

  // A fragments (ISA 16x4 f32 layout): lane<16 K=4t,4t+1 ; lane>=16 K=4t+2,4t+3
  v2f a[64];
  const float* xrow = x + (mbase + lrow) * D_DIM + khalf;
#pragma unroll
  for (int t = 0; t < 64; ++t)
    a[t] = *(const v2f*)(xrow + 4 * t);

  float minv[8];
  int   mini[8];
#pragma unroll
  for (int v = 0; v < 8; ++v) { minv[v] = __builtin_inff(); mini[v] = 0; }

  // prologue: fetch tile 0
  if (wave == 0) {
    tdm_load_tile(cent + (size_t)nbase0 * D_DIM, 0u);
    __builtin_amdgcn_s_wait_tensorcnt(0);
  }
  __syncthreads();
  LDS_PUBLISH(smem);                   // tile 0 visible to the compiler

  const float* bbase = smem + lrow * TROW + khalf;

  for (int i = 0; i < NITER; ++i) {
    const int cur = i & 1;
    // overlap: DMA tile i+1 while computing tile i
    if (wave == 0 && (i + 1) < NITER)
      tdm_load_tile(cent + (size_t)(nbase0 + (i + 1) * 16) * D_DIM,
                    (unsigned)((cur ^ 1) * TILE_B));

    const float* bp = bbase + cur * TILE_F;
    v8f c = {0.f, 0.f, 0.f, 0.f, 0.f, 0.f, 0.f, 0.f};
    float sqa[4] = {0.f, 0.f, 0.f, 0.f};          // 4 chains: no serial FMA bottleneck

    // software-pipelined B fragments: PF loads in flight hide LDS latency
    v2f breg[PF];
#pragma unroll
    for (int t = 0; t < PF; ++t)
      breg[t] = *(const v2f*)(bp + 4 * t);
#pragma unroll
    for (int t = 0; t < 64; ++t) {
      v2f b = breg[t % PF];
      if (t + PF < 64)
        breg[t % PF] = *(const v2f*)(bp + 4 * (t + PF));
      sqa[t & 3] = __builtin_fmaf(b.x, b.x, sqa[t & 3]);
      sqa[t & 3] = __builtin_fmaf(b.y, b.y, sqa[t & 3]);
      c = __builtin_amdgcn_wmma_f32_16x16x4_f32(false, a[t], false, b,
                                                (short)0, c, false, false);
    }
    float sq = (sqa[0] + sqa[1]) + (sqa[2] + sqa[3]);
    float csq = sq + __shfl_xor(sq, 16, 32);      // join the two K halves
    const int n = nbase0 + i * 16 + lrow;         // column this lane tracks
#pragma unroll
    for (int v = 0; v < 8; ++v) {
      float dist = __builtin_fmaf(-2.f, c[v], csq);
      if (dist < minv[v]) { minv[v] = dist; mini[v] = n; }
    }

    if (wave == 0) __builtin_amdgcn_s_wait_tensorcnt(0);
    __syncthreads();                              // tile i+1 ready / buffer reusable
    LDS_PUBLISH(smem);                            // new tile visible to the compiler
  }

  // Tree-reduce (min, argmin) across the 16 lanes sharing each row.
  // VGPR v: lanes 0-15 -> row mbase+v ; lanes 16-31 -> row mbase+v+8.
#pragma unroll
  for (int v = 0; v < 8; ++v) {
    float mv = minv[v]; int mi = mini[v];
#pragma unroll
    for (int s = 8; s >= 1; s >>= 1) {
      float ov = __shfl_xor(mv, s, 32);
      int   oi = __shfl_xor(mi, s, 32);
      if (ov < mv || (ov == mv && oi < mi)) { mv = ov; mi = oi; }
    }
    if (lrow == 0) {
      const int m = mbase + v + ((lane >> 4) ? 8 : 0);
      float2 r; r.x = mv; r.y = __int_as_float(mi);
      part[m * NSPLIT + nspl] = r;
    }
  }
}

// ---------------------------------------------------------------------------
// Kernel 2: reduce NSPLIT partials per row, emit code, gather centroid row.
// One wave per output row; coalesced float4 gather (1 KB per row).
// ---------------------------------------------------------------------------
__global__ __launch_bounds__(256) void vq_reduce_gather(
    const float* __restrict__ cent,
    const float2* __restrict__ part,
    float* __restrict__ out_x,        // [4096*256]
    int* __restrict__ out_codes)      // [4096]
{
  const int lane = threadIdx.x & 31;
  const int wave = threadIdx.x >> 5;
  const int m = blockIdx.x * 8 + wave;

  float mv = __builtin_inff(); int mi = 0;
  if (lane < NSPLIT) {
    float2 r = part[m * NSPLIT + lane];
    mv = r.x; mi = __float_as_int(r.y);
  }
#pragma unroll
  for (int s = 8; s >= 1; s >>= 1) {
    float ov = __shfl_xor(mv, s, 32);
    int   oi = __shfl_xor(mi, s, 32);
    if (ov < mv || (ov == mv && oi < mi)) { mv = ov; mi = oi; }
  }
  const int code = __shfl(mi, 0, 32);

  const float4* src = (const float4*)(cent + (size_t)code * D_DIM);
  float4* dst = (float4*)(out_x + (size_t)m * D_DIM);
  dst[lane * 2]     = src[lane * 2];
  dst[lane * 2 + 1] = src[lane * 2 + 1];
  if (lane == 0) out_codes[m] = code;
}

// ---------------------------------------------------------------------------
extern "C" void kernel_launch(void* const* d_in, const int* in_sizes, int n_in,
                              void* d_out, int out_size, void* d_ws, size_t ws_size,
                              hipStream_t stream) {
  // d_in order: xhat_BFD (unused, shape only), x_BD, centroids
  const float* x    = (const float*)d_in[1];
  const float* cent = (const float*)d_in[2];
  float*  out   = (float*)d_out;
  float2* part  = (float2*)d_ws;     // 4096 * NSPLIT * 8 B = 512 KB
  int*    codes = (int*)(out + (size_t)M_TOT * D_DIM);

  dim3 g1(NSPLIT, M_TOT / (16 * WPB));             // (16, 64) blocks, 4 waves each
  vq_argmin_partial<<<g1, WPB * 32, 0, stream>>>(x, cent, part);

  vq_reduce_gather<<<M_TOT / 8, 256, 0, stream>>>(cent, part, out, codes);
}